// RelativePositionalEmbedding2DKey_57887569215532
// MI455X (gfx1250) — compile-verified
//
#include <hip/hip_runtime.h>

// ---------------------------------------------------------------------------
// RelativePositionalEmbedding2DKey for MI455X (gfx1250, wave32).
//
// Static config: B=2, N=8, T=8, H=W=14, C=64.
//   rel_h[bn,q,s,h,w] = sum_c embH[h-q+13, c] * key[bn,s,h,w,c]   (q in 0..13)
//   rel_w[bn,q,s,h,w] = sum_c embW[w-q+13, c] * key[bn,s,h,w,c]   (q in 0..13)
//   rel_t[bn,q,s,h,w] = sum_c embT[s-q+ 7, c] * key[bn,s,h,w,c]   (q in 0..7)
//   out[bn, (tq,qh,qw), p] = scores + rel_h[qh,p] + rel_w[qw,p] + rel_t[tq,p]
//
// Kernel 1: V_WMMA_F32_16X16X4_F32 tiles (full f32 precision, matches ref).
// Kernel 2: bandwidth-bound float4 streaming add (~315 MB => ~13.5 us floor).
// ---------------------------------------------------------------------------

typedef __attribute__((ext_vector_type(2))) float v2f;
typedef __attribute__((ext_vector_type(8))) float v8f;

#define BN    16      // B*N
#define CDIM  64
#define PDIM  1568    // 8*14*14 key positions
#define NQ    1568    // 8*14*14 query positions

#define JOBS_H (BN * 14 * 7)    // 1568: (bn, h, tile of 16 over 112 (s,w))
#define JOBS_W (BN * 14 * 7)    // 1568: (bn, w, tile of 16 over 112 (s,h))
#define JOBS_T (BN * 8 * 13)    // 1664: (bn, s, tile of 16 over 196 (h,w))
#define JOBS_TOTAL (JOBS_H + JOBS_W + JOBS_T)   // 4800

__global__ __launch_bounds__(32) void rel_gemm_wmma(
    const float* __restrict__ key,    // [BN, 8,14,14, 64]
    const float* __restrict__ embH,   // [27, 64]
    const float* __restrict__ embW,   // [27, 64]
    const float* __restrict__ embT,   // [15, 64]
    float* __restrict__ relH,         // [BN, 14, 1568]
    float* __restrict__ relW,         // [BN, 14, 1568]
    float* __restrict__ relT)         // [BN,  8, 1568]
{
    int job = blockIdx.x;
    const int lane = threadIdx.x;     // one wave32 per block
    const int half = lane >> 4;       // 0: K={0,1} / M rows 0-7 ; 1: K={2,3} / M rows 8-15
    const int col  = lane & 15;

    int mode, bn, fix, tile;
    if (job < JOBS_H) {
        mode = 0;
        bn = job / 98; int r = job % 98; fix = r / 7; tile = r % 7;
    } else if (job < JOBS_H + JOBS_W) {
        mode = 1; job -= JOBS_H;
        bn = job / 98; int r = job % 98; fix = r / 7; tile = r % 7;
    } else {
        mode = 2; job -= (JOBS_H + JOBS_W);
        bn = job / 104; int r = job % 104; fix = r / 13; tile = r % 13;
    }

    const int  Jmax = (mode == 2) ? 196 : 112;   // columns in this slice
    const int  QM   = (mode == 2) ? 8 : 14;      // valid query rows
    const int  off  = (mode == 2) ? 7 : 13;      // rel-index offset
    const int  rows = (mode == 2) ? 15 : 27;     // embedding table rows
    const int  Qdim = QM;
    const float* emb = (mode == 0) ? embH : (mode == 1) ? embW : embT;
    float* outp      = (mode == 0) ? relH : (mode == 1) ? relW : relT;

    // ---- B operand: key column for position j (clamped in-bounds; unused
    //      columns are simply never stored) ----
    const int j  = tile * 16 + col;
    const int jc = (j < Jmax) ? j : 0;
    int p;
    if (mode == 2) {
        p = fix * 196 + jc;                       // fixed s, j = h*14+w
    } else {
        int s = jc / 14, o = jc % 14;
        p = (mode == 1) ? (s * 196 + o * 14 + fix)   // fixed w, j = s*14+h
                        : (s * 196 + fix * 14 + o);  // fixed h, j = s*14+w
    }
    const float* kp = key + ((size_t)bn * PDIM + p) * CDIM;

    // ---- A operand: embedding row for query q = col (clamped; rows q>=QM
    //      land in D rows we never store) ----
    int idx = fix - col + off;
    idx = idx < 0 ? 0 : (idx > rows - 1 ? rows - 1 : idx);
    const float* arow = emb + (size_t)idx * CDIM;

    // ---- D = A(16x64) x B(64x16) via 16 chained V_WMMA_F32_16X16X4_F32 ----
    v8f acc = {};
#pragma unroll
    for (int kk = 0; kk < 16; ++kk) {
        const int k0 = kk * 4 + 2 * half;  // lanes 0-15: K=k0,k0+1 ; 16-31: K=k0+2,k0+3
        v2f a, b;
        a.x = arow[k0]; a.y = arow[k0 + 1];
        b.x = kp[k0];   b.y = kp[k0 + 1];
        acc = __builtin_amdgcn_wmma_f32_16x16x4_f32(
            /*neg_a=*/false, a, /*neg_b=*/false, b,
            /*c_mod=*/(short)0, acc, /*reuse_a=*/false, /*reuse_b=*/false);
    }

    // ---- store valid tile region: lane half selects rows 0-7 vs 8-15 ----
    if (j < Jmax) {
#pragma unroll
        for (int r = 0; r < 8; ++r) {
            const int q = r + 8 * half;
            if (q < QM) {
                outp[((size_t)bn * Qdim + q) * PDIM + p] = acc[r];
            }
        }
    }
}

// ---------------------------------------------------------------------------
// Streaming add: one block per output row (bn, qidx); 392 float4 per row.
// ---------------------------------------------------------------------------
__global__ __launch_bounds__(256) void add_rel(
    const float* __restrict__ scores,
    const float* __restrict__ relH,
    const float* __restrict__ relW,
    const float* __restrict__ relT,
    float* __restrict__ out)
{
    const int row = blockIdx.x;            // bn*1568 + qidx
    const int bn  = row / NQ;
    const int q   = row - bn * NQ;
    const int tq  = q / 196;
    const int rem = q - tq * 196;
    const int qh  = rem / 14;
    const int qw  = rem - qh * 14;

    const float4* s = (const float4*)(scores + (size_t)row * PDIM);
    const float4* h = (const float4*)(relH + ((size_t)bn * 14 + qh) * PDIM);
    const float4* w = (const float4*)(relW + ((size_t)bn * 14 + qw) * PDIM);
    const float4* t = (const float4*)(relT + ((size_t)bn * 8  + tq) * PDIM);
    float4*       o = (float4*)(out + (size_t)row * PDIM);

    for (int i = threadIdx.x; i < PDIM / 4; i += 256) {
        float4 sv = s[i], hv = h[i], wv = w[i], tv = t[i];
        float4 r;
        r.x = sv.x + hv.x + wv.x + tv.x;
        r.y = sv.y + hv.y + wv.y + tv.y;
        r.z = sv.z + hv.z + wv.z + tv.z;
        r.w = sv.w + hv.w + wv.w + tv.w;
        o[i] = r;
    }
}

extern "C" void kernel_launch(void* const* d_in, const int* in_sizes, int n_in,
                              void* d_out, int out_size, void* d_ws, size_t ws_size,
                              hipStream_t stream) {
    const float* key    = (const float*)d_in[0];  // [2,8,8,14,14,64]
    const float* scores = (const float*)d_in[1];  // [2,8,1568,1568]
    const float* embH   = (const float*)d_in[2];  // [27,64]
    const float* embW   = (const float*)d_in[3];  // [27,64]
    const float* embT   = (const float*)d_in[4];  // [15,64]

    float* relH = (float*)d_ws;                   // 16*14*1568 floats
    float* relW = relH + (size_t)BN * 14 * PDIM;  // 16*14*1568 floats
    float* relT = relW + (size_t)BN * 14 * PDIM;  // 16* 8*1568 floats
    // total workspace: 3,612,672 bytes

    rel_gemm_wmma<<<JOBS_TOTAL, 32, 0, stream>>>(key, embH, embW, embT,
                                                 relH, relW, relT);
    add_rel<<<BN * NQ, 256, 0, stream>>>(scores, relH, relW, relT, (float*)d_out);
}